// SpikeFP32LayerNorm_11450382811502
// MI455X (gfx1250) — compile-verified
//
#include <hip/hip_runtime.h>

// SpikeFP32LayerNorm for MI455X (gfx1250).
// Streaming, single-pass: ~805 MB total traffic -> ~35 us floor at 23.3 TB/s.
// gfx1250 async global->LDS DMA (global_load_async_to_lds_b128) double-buffered
// against decode; s_wait_asynccnt with nonzero threshold exploits in-order
// completion of a wave's async loads. ds_load_b128 decode from padded LDS.

#define N_VALS        768          // values per row (layernorm axis)
#define BITS          32           // pulses per value
#define BLOCK         256          // 8 wave32
#define PHASES        3            // 768 / 256
#define VAL_STRIDE_B  144          // 128B payload + 16B pad: 16B-aligned, 2-pass LDS banks
#define VAL_STRIDE_F  (VAL_STRIDE_B / 4)
#define PHASE_BYTES   (BLOCK * VAL_STRIDE_B)   // 36,864 B per staging buffer
#define ROW_FLOATS    (N_VALS * BITS)          // 24576 floats = 96 KB per row

__device__ __forceinline__ void issue_phase_async(const float* __restrict__ grow,
                                                  unsigned lds_base, int phase,
                                                  int buf, int tid) {
    const float* gphase = grow + phase * (BLOCK * BITS);
    const unsigned buf_base = lds_base + (unsigned)(buf * PHASE_BYTES);
#pragma unroll
    for (int k = 0; k < 8; ++k) {
        const int      j    = tid + k * BLOCK;       // 16B chunk id in phase
        const int      v    = j >> 3;                // local value index
        const int      c    = j & 7;                 // 16B chunk within value
        const unsigned loff = buf_base + (unsigned)(v * VAL_STRIDE_B + c * 16);
        const void*    g    = (const void*)(gphase + j * 4);
        asm volatile("global_load_async_to_lds_b128 %0, %1, off"
                     :: "v"(loff), "v"(g) : "memory");
    }
}

// 32 MSB-first pulses -> u32, as 4 independent 8-bit chains for VALU ILP.
__device__ __forceinline__ unsigned decode_value(const float* __restrict__ base) {
    const float4* bp = (const float4*)base;
    unsigned acc[4];
#pragma unroll
    for (int b = 0; b < 4; ++b) {
        float4 q0 = bp[2 * b];
        float4 q1 = bp[2 * b + 1];
        unsigned a = 0u;
        a = (a << 1) | ((__float_as_uint(q0.x) >> 23) & 1u);
        a = (a << 1) | ((__float_as_uint(q0.y) >> 23) & 1u);
        a = (a << 1) | ((__float_as_uint(q0.z) >> 23) & 1u);
        a = (a << 1) | ((__float_as_uint(q0.w) >> 23) & 1u);
        a = (a << 1) | ((__float_as_uint(q1.x) >> 23) & 1u);
        a = (a << 1) | ((__float_as_uint(q1.y) >> 23) & 1u);
        a = (a << 1) | ((__float_as_uint(q1.z) >> 23) & 1u);
        a = (a << 1) | ((__float_as_uint(q1.w) >> 23) & 1u);
        acc[b] = a;
    }
    return (acc[0] << 24) | (acc[1] << 16) | (acc[2] << 8) | acc[3];
}

__global__ __launch_bounds__(BLOCK)
void spike_fp32_layernorm_kernel(const float* __restrict__ xin,
                                 float* __restrict__ xout) {
#pragma clang fp contract(off)
    __shared__ float  smem[2 * BLOCK * VAL_STRIDE_F];   // 73,728 B double buffer
    __shared__ double red[BLOCK];                       // 2 KB reduction

    const int       tid  = threadIdx.x;
    const long long row  = blockIdx.x;
    const float* __restrict__ grow = xin  + row * (long long)ROW_FLOATS;
    float*       __restrict__ orow = xout + row * (long long)ROW_FLOATS;

    // LDS byte offset of staging buffer (low 32 bits of generic pointer).
    const unsigned lds_base = (unsigned)(size_t)(&smem[0]);

    double xv[PHASES];

    // ---- Software-pipelined: DMA phase p+1 while decoding phase p ----
    issue_phase_async(grow, lds_base, 0, 0, tid);
    issue_phase_async(grow, lds_base, 1, 1, tid);

    // phase 0 (buffer 0): async loads complete in order -> cnt<=8 means phase0 landed
    asm volatile("s_wait_asynccnt 0x8" ::: "memory");
    __syncthreads();
    xv[0] = (double)__uint_as_float(decode_value(&smem[tid * VAL_STRIDE_F]));
    __syncthreads();                     // buffer 0 free for phase 2

    issue_phase_async(grow, lds_base, 2, 0, tid);

    // phase 1 (buffer 1)
    asm volatile("s_wait_asynccnt 0x8" ::: "memory");
    __syncthreads();
    xv[1] = (double)__uint_as_float(
        decode_value(&smem[BLOCK * VAL_STRIDE_F + tid * VAL_STRIDE_F]));
    __syncthreads();

    // phase 2 (buffer 0)
    asm volatile("s_wait_asynccnt 0x0" ::: "memory");
    __syncthreads();
    xv[2] = (double)__uint_as_float(decode_value(&smem[tid * VAL_STRIDE_F]));
    __syncthreads();

    // ---- Reduction 1: mean (FP64) ----
    red[tid] = xv[0] + xv[1] + xv[2];
    __syncthreads();
    for (int off = BLOCK / 2; off > 0; off >>= 1) {
        if (tid < off) red[tid] += red[tid + off];
        __syncthreads();
    }
    const double mean = red[0] / 768.0;
    __syncthreads();

    // ---- Reduction 2: centered variance (FP64, two-pass like reference) ----
    const double xc0 = xv[0] - mean;
    const double xc1 = xv[1] - mean;
    const double xc2 = xv[2] - mean;
    red[tid] = xc0 * xc0 + xc1 * xc1 + xc2 * xc2;
    __syncthreads();
    for (int off = BLOCK / 2; off > 0; off >>= 1) {
        if (tid < off) red[tid] += red[tid + off];
        __syncthreads();
    }
    const double var = red[0] / 768.0;

    // ---- Newton-Raphson rsqrt, seeded with exact 1/a (matches reference) ----
    const double a = var + 1e-6;
    double y = 1.0 / a;
#pragma unroll
    for (int it = 0; it < 5; ++it) {
        const double t = y * y;
        const double w = 3.0 - a * t;
        y = 0.5 * (y * w);
    }

    // ---- Normalize, round to FP32, re-encode 32 MSB-first pulses, store ----
    const double xc[PHASES] = { xc0, xc1, xc2 };
#pragma unroll
    for (int p = 0; p < PHASES; ++p) {
        const float    f = (float)(xc[p] * y);
        const unsigned u = __float_as_uint(f);
        float4* op = (float4*)(orow + (long long)(p * BLOCK + tid) * BITS);
#pragma unroll
        for (int c = 0; c < 8; ++c) {
            float4 q;
            q.x = (float)((u >> (31 - 4 * c)) & 1u);
            q.y = (float)((u >> (30 - 4 * c)) & 1u);
            q.z = (float)((u >> (29 - 4 * c)) & 1u);
            q.w = (float)((u >> (28 - 4 * c)) & 1u);
            op[c] = q;
        }
    }
}

extern "C" void kernel_launch(void* const* d_in, const int* in_sizes, int n_in,
                              void* d_out, int out_size, void* d_ws, size_t ws_size,
                              hipStream_t stream) {
    (void)n_in; (void)out_size; (void)d_ws; (void)ws_size;
    const float* x   = (const float*)d_in[0];
    float*       out = (float*)d_out;
    const long long n    = (long long)in_sizes[0];
    const int       rows = (int)(n / (long long)ROW_FLOATS);   // 4096 for [8,512,768,32]
    spike_fp32_layernorm_kernel<<<rows, BLOCK, 0, stream>>>(x, out);
}